// JK_86672440033909
// MI455X (gfx1250) — compile-verified
//
#include <hip/hip_runtime.h>

typedef float v2f __attribute__((ext_vector_type(2)));
typedef float v8f __attribute__((ext_vector_type(8)));

#define N_NODES  100000
#define N_EDGES  1600000
#define N_GRAPHS 128
#define F_IN     128
#define F_HID    64
#define JKW      192   // 3 * F_HID
#define LINW     128
#define OUTW     10

// ---------------- degree / normalization ----------------

__global__ void deg_kernel(const int* __restrict__ dst, float* __restrict__ deg) {
    int e = blockIdx.x * blockDim.x + threadIdx.x;
    if (e < N_EDGES) atomicAdd(&deg[dst[e]], 1.0f);
}

__global__ void dinv_kernel(const float* __restrict__ deg, float* __restrict__ dinv) {
    int v = blockIdx.x * blockDim.x + threadIdx.x;
    if (v < N_NODES) dinv[v] = rsqrtf(deg[v] + 1.0f);   // +1 accounts for self-loop
}

// ---------------- dense GEMM via V_WMMA_F32_16X16X4_F32 ----------------
// H[N x 64] = X[N x K, row stride lda] @ W[K x 64]
// 128 threads = 4 waves; block computes a 16x64 tile; each wave one 16x16 tile.
__global__ __launch_bounds__(128) void gemm_wmma(const float* __restrict__ X, int lda,
                                                 const float* __restrict__ W, int K,
                                                 float* __restrict__ H) {
    __shared__ float sW[128 * 64];                     // 32 KB (max K = 128)
    for (int i = threadIdx.x; i < K * 64; i += 128) sW[i] = W[i];
    __syncthreads();

    const int lane    = threadIdx.x & 31;
    const int wave    = threadIdx.x >> 5;              // output cols [16*wave, 16*wave+16)
    const int rowbase = blockIdx.x * 16;
    const int half    = lane >> 4;                     // 0: K pair (0,1); 1: K pair (2,3)
    const int lr      = lane & 15;

    // A: lane (half,lr) holds X[rowbase+lr][kb + 2*half .. +1]  (aligned float2)
    const float* xr = X + (size_t)(rowbase + lr) * lda + half * 2;
    // B: lane (half,lr) holds W[kb + 2*half][col], W[kb + 2*half + 1][col]
    const float* wb = sW + (half * 2) * 64 + wave * 16 + lr;

    v8f c = {};
    #pragma unroll 4
    for (int kb = 0; kb < K; kb += 4) {
        v2f a = *(const v2f*)(xr + kb);
        v2f b;
        b.x = wb[kb * 64];
        b.y = wb[kb * 64 + 64];
        // D = A(16x4,f32) * B(4x16,f32) + C  -> v_wmma_f32_16x16x4_f32
        c = __builtin_amdgcn_wmma_f32_16x16x4_f32(false, a, false, b,
                                                  (short)0, c, false, false);
    }

    // C/D layout: VGPR i -> row (rowbase + i + 8*half), col (16*wave + lr)
    float* hr = H + (size_t)(rowbase + half * 8) * 64 + wave * 16 + lr;
    #pragma unroll
    for (int i = 0; i < 8; ++i) hr[(size_t)i * 64] = c[i];
}

// ---------------- edge aggregation: agg[v] += dinv[u]*dinv[v]*h[u] ----------------
// One thread per (edge, feature-pair): 32 threads cover one 64-float row (coalesced).
__global__ void aggregate_kernel(const int* __restrict__ src, const int* __restrict__ dst,
                                 const float* __restrict__ dinv,
                                 const float* __restrict__ h, float* __restrict__ agg) {
    long long gid = (long long)blockIdx.x * blockDim.x + threadIdx.x;
    int e = (int)(gid >> 5);
    int p = ((int)gid & 31) * 2;
    if (e < N_EDGES) {
        int u = src[e], v = dst[e];
        float w = dinv[u] * dinv[v];
        v2f hv = *(const v2f*)(h + (size_t)u * 64 + p);
        atomicAdd(&agg[(size_t)v * 64 + p],     w * hv.x);
        atomicAdd(&agg[(size_t)v * 64 + p + 1], w * hv.y);
    }
}

// ---------------- self-loop + bias + ReLU, write into JK concat block ----------------
__global__ void bias_relu_kernel(const float* __restrict__ agg, const float* __restrict__ h,
                                 const float* __restrict__ dinv, const float* __restrict__ b,
                                 float* __restrict__ hjk, int off) {
    int gid = blockIdx.x * blockDim.x + threadIdx.x;
    if (gid < N_NODES * 64) {
        int v = gid >> 6, f = gid & 63;
        float di  = dinv[v];
        float val = agg[gid] + di * di * h[gid] + b[f];     // self-loop norm = dinv^2
        hjk[(size_t)v * JKW + off + f] = fmaxf(val, 0.0f);
    }
}

// ---------------- global max pool (values >= 0 so int-bit atomicMax is monotone) ----
__global__ void pool_kernel(const float* __restrict__ hjk, const int* __restrict__ batch,
                            float* __restrict__ pooled) {
    int gid = blockIdx.x * blockDim.x + threadIdx.x;
    if (gid < N_NODES * JKW) {
        int v = gid / JKW, c = gid - v * JKW;
        int g = batch[v];
        atomicMax((int*)&pooled[g * JKW + c], __float_as_int(hjk[gid]));
    }
}

// ---------------- tiny dense heads ----------------
__global__ void lin_kernel(const float* __restrict__ pooled, const float* __restrict__ Wl,
                           const float* __restrict__ bl, float* __restrict__ z) {
    int gid = blockIdx.x * blockDim.x + threadIdx.x;   // 128*128
    if (gid < N_GRAPHS * LINW) {
        int g = gid >> 7, j = gid & 127;
        float acc = bl[j];
        #pragma unroll 4
        for (int k = 0; k < JKW; ++k) acc += pooled[g * JKW + k] * Wl[k * LINW + j];
        z[gid] = acc;
    }
}

__global__ void out_kernel(const float* __restrict__ z, const float* __restrict__ Wo,
                           const float* __restrict__ bo, float* __restrict__ out) {
    int gid = blockIdx.x * blockDim.x + threadIdx.x;   // 128*10
    if (gid < N_GRAPHS * OUTW) {
        int g = gid / OUTW, o = gid - g * OUTW;
        float acc = bo[o];
        #pragma unroll 4
        for (int k = 0; k < LINW; ++k) acc += z[g * LINW + k] * Wo[k * OUTW + o];
        out[gid] = acc;
    }
}

// ---------------- host orchestration ----------------
extern "C" void kernel_launch(void* const* d_in, const int* in_sizes, int n_in,
                              void* d_out, int out_size, void* d_ws, size_t ws_size,
                              hipStream_t stream) {
    const float* x     = (const float*)d_in[0];
    const int*   eidx  = (const int*)d_in[1];          // [2, E] row-major
    const int*   batch = (const int*)d_in[2];
    const float* W1 = (const float*)d_in[3];  const float* b1 = (const float*)d_in[4];
    const float* W2 = (const float*)d_in[5];  const float* b2 = (const float*)d_in[6];
    const float* W3 = (const float*)d_in[7];  const float* b3 = (const float*)d_in[8];
    const float* lin_W = (const float*)d_in[9];  const float* lin_b = (const float*)d_in[10];
    const float* out_W = (const float*)d_in[11]; const float* out_b = (const float*)d_in[12];
    float* out = (float*)d_out;

    const int* src = eidx;
    const int* dst = eidx + N_EDGES;

    // workspace layout
    char* ws = (char*)d_ws;
    float* deg    = (float*)ws;                           ws += (size_t)N_NODES * 4;
    float* dinv   = (float*)ws;                           ws += (size_t)N_NODES * 4;
    float* h_tmp  = (float*)ws;                           ws += (size_t)N_NODES * 64 * 4;
    float* agg    = (float*)ws;                           ws += (size_t)N_NODES * 64 * 4;
    float* hjk    = (float*)ws;                           ws += (size_t)N_NODES * JKW * 4;
    float* pooled = (float*)ws;                           ws += (size_t)N_GRAPHS * JKW * 4;
    float* z      = (float*)ws;                           ws += (size_t)N_GRAPHS * LINW * 4;

    const int T = 256;

    // degrees + normalization
    hipMemsetAsync(deg, 0, (size_t)N_NODES * 4, stream);
    deg_kernel<<<(N_EDGES + T - 1) / T, T, 0, stream>>>(dst, deg);
    dinv_kernel<<<(N_NODES + T - 1) / T, T, 0, stream>>>(deg, dinv);

    const long long eth = (long long)N_EDGES * 32;       // aggregation threads
    const int eg = (int)((eth + T - 1) / T);
    const int ng64 = (N_NODES * 64 + T - 1) / T;

    const float* layer_in[3] = { x, hjk, hjk + 64 };
    const int    layer_lda[3] = { F_IN, JKW, JKW };
    const int    layer_K[3]   = { F_IN, F_HID, F_HID };
    const float* layer_W[3] = { W1, W2, W3 };
    const float* layer_b[3] = { b1, b2, b3 };

    for (int l = 0; l < 3; ++l) {
        gemm_wmma<<<N_NODES / 16, 128, 0, stream>>>(layer_in[l], layer_lda[l],
                                                    layer_W[l], layer_K[l], h_tmp);
        hipMemsetAsync(agg, 0, (size_t)N_NODES * 64 * 4, stream);
        aggregate_kernel<<<eg, T, 0, stream>>>(src, dst, dinv, h_tmp, agg);
        bias_relu_kernel<<<ng64, T, 0, stream>>>(agg, h_tmp, dinv, layer_b[l],
                                                 hjk, l * F_HID);
    }

    // global max pool over batch
    hipMemsetAsync(pooled, 0, (size_t)N_GRAPHS * JKW * 4, stream);
    pool_kernel<<<(N_NODES * JKW + T - 1) / T, T, 0, stream>>>(hjk, batch, pooled);

    // heads
    lin_kernel<<<(N_GRAPHS * LINW + T - 1) / T, T, 0, stream>>>(pooled, lin_W, lin_b, z);
    out_kernel<<<(N_GRAPHS * OUTW + T - 1) / T, T, 0, stream>>>(z, out_W, out_b, out);
}